// DCT_Layer_new_7851200217575
// MI455X (gfx1250) — compile-verified
//
#include <hip/hip_runtime.h>

typedef __attribute__((ext_vector_type(2))) float v2f;
typedef __attribute__((ext_vector_type(8))) float v8f;

namespace {
constexpr int Bn = 8, Cn = 3, H = 512, W = 512, OH = 513, OW = 513;
constexpr int XT     = 11;              // 16-wide x-subtiles per block
constexpr int TILE_X = 16 * XT;         // 176 output cols per block
constexpr int TILE_Y = 8;               // 8 output rows per block (1/wave)
constexpr int LDS_H  = TILE_Y + 4;      // 12 input rows (2-halo each side)
constexpr int LDS_W  = TILE_X + 4;      // 180 input cols
}

// 1-D DCT-II factors u_k * cos(pi/8 * k * (2i+1)), row-major [k][i].
__constant__ float DCT1D[16] = {
    0.50000000f,  0.50000000f,  0.50000000f,  0.50000000f,
    0.65328148f,  0.27059805f, -0.27059805f, -0.65328148f,
    0.50000000f, -0.50000000f, -0.50000000f,  0.50000000f,
    0.27059805f, -0.65328148f,  0.65328148f, -0.27059805f,
};

__global__ __launch_bounds__(256)
void dct4x4_wmma_kernel(const float* __restrict__ x, float* __restrict__ out)
{
    __shared__ float tile[LDS_H * LDS_W];

    const int tid  = threadIdx.x;
    const int lane = tid & 31;
    const int wv   = tid >> 5;          // wave id = output row within tile
    const int n    = lane & 15;         // WMMA N column / A-matrix row
    const int half = lane >> 4;         // 0: K rows {0,1}; 1: K rows {2,3}

    const int plane = blockIdx.z;       // b*Cn + c
    const int x0 = blockIdx.x * TILE_X;
    const int y0 = blockIdx.y * TILE_Y;

    // ---- stage zero-padded input tile into LDS (loop fill, coalesced) ----
    const float* xin = x + (size_t)plane * H * W;
    for (int idx = tid; idx < LDS_H * LDS_W; idx += 256) {
        int r  = idx / LDS_W;
        int cc = idx - r * LDS_W;
        int gy = y0 - 2 + r;
        int gx = x0 - 2 + cc;
        float v = 0.0f;
        if ((unsigned)gy < (unsigned)H && (unsigned)gx < (unsigned)W)
            v = xin[gy * W + gx];
        tile[idx] = v;
    }
    __syncthreads();

    // ---- A-matrix (16 filters x 4 taps per chunk), hoisted out of the loop.
    // A[m, k], m = kk*4+ll, chunk i covers taps k = i*4 + j, j = 0..3.
    // Lane layout: lanes 0-15 hold K={0,1} in v[0],v[1]; lanes 16-31 K={2,3}.
    const int kk = n >> 2;
    const int ll = n & 3;
    const int j0 = half * 2;
    const float c0 = DCT1D[ll * 4 + j0];
    const float c1 = DCT1D[ll * 4 + j0 + 1];
    v2f a[4];
#pragma unroll
    for (int i = 0; i < 4; ++i) {
        float rf = DCT1D[kk * 4 + i];
        a[i][0] = rf * c0;
        a[i][1] = rf * c1;
    }

    const int  y    = y0 + wv;
    const bool yok  = (y < OH);
    const size_t fstride = (size_t)OH * OW;
    // Base for this lane's store column; filter plane r*fstride and 16*t fit
    // the 24-bit store immediate, so all stores share this one address.
    float* obase = out + ((size_t)plane * 16 + (size_t)half * 8) * fstride
                       + (size_t)y * OW + x0 + n;
    const float* brow = &tile[wv * LDS_W + n + j0];

    for (int t = 0; t < XT; ++t) {
        // ---- 4 chained V_WMMA_F32_16X16X4_F32: D = A(16x4)*B(4x16) + D ----
        v8f acc = {};
#pragma unroll
        for (int i = 0; i < 4; ++i) {
            v2f b;
            b[0] = brow[i * LDS_W + 16 * t];
            b[1] = brow[i * LDS_W + 16 * t + 1];
            acc = __builtin_amdgcn_wmma_f32_16x16x4_f32(
                    /*neg_a=*/false, a[i], /*neg_b=*/false, b,
                    /*c_mod=*/(short)0, acc, /*reuse_a=*/false, /*reuse_b=*/false);
        }

        // ---- DCT_Trunc epilogue + non-temporal coalesced stores ----
        // C/D layout: VGPR r -> filter M = r + 8*half, pixel N = lane%16.
        const int xc = x0 + 16 * t + n;
        if (yok && xc < OW) {
#pragma unroll
            for (int r = 0; r < 8; ++r) {
                float v = fminf(fabsf(acc[r]), 8.0f);
                __builtin_nontemporal_store(v, &obase[16 * t + (size_t)r * fstride]);
            }
        }
    }
}

extern "C" void kernel_launch(void* const* d_in, const int* in_sizes, int n_in,
                              void* d_out, int out_size, void* d_ws, size_t ws_size,
                              hipStream_t stream) {
    (void)in_sizes; (void)n_in; (void)d_ws; (void)ws_size; (void)out_size;
    const float* xin = (const float*)d_in[0];
    float* out = (float*)d_out;
    dim3 grid((OW + TILE_X - 1) / TILE_X,   // 3
              (OH + TILE_Y - 1) / TILE_Y,   // 65
              Bn * Cn);                     // 24
    dct4x4_wmma_kernel<<<grid, dim3(256), 0, stream>>>(xin, out);
}